// MPN_solvation_attention_33148557590930
// MI455X (gfx1250) — compile-verified
//
#include <hip/hip_runtime.h>
#include <hip/hip_bf16.h>
#include <math.h>

// ---------------------------------------------------------------------------
// Problem constants (from the reference)
// ---------------------------------------------------------------------------
#define H 512
#define MAX_NB 6
#define NB 64
#define A_SV 24
#define A_SU 32
#define ATOM_FDIM 133
#define BOND_FDIM 147
#define N_A_SV (NB * A_SV)          // 1536
#define N_B_SV (2 * N_A_SV)         // 3072
#define N_A_SU (NB * A_SU)          // 2048
#define N_B_SU (2 * N_A_SU)         // 4096
#define NPAIR (NB * A_SV * A_SU)    // 49152

typedef float v2f __attribute__((ext_vector_type(2)));
typedef float v8f __attribute__((ext_vector_type(8)));

// ---------------------------------------------------------------------------
// Generic fp32 WMMA GEMM:  D = act( A[M,K] @ W[K,N] + bias[N] + Cadd[M,N] )
// One wave computes a 32(M) x 64(N) tile via V_WMMA_F32_16X16X4_F32:
// two A fragments (rows m0..m0+15 and m0+16..m0+31) share four B fragments.
// Main K loop is branch-free; the K%4 tail (K=147/133) uses clamped-index
// loads + select so EXEC stays all-ones (required by WMMA) with no branches.
// Requirements: M%32==0, N%64==0. K arbitrary.
// ---------------------------------------------------------------------------
__global__ __launch_bounds__(256)
void gemm_wmma_f32(const float* __restrict__ A, int lda,
                   const float* __restrict__ W, int ldw,
                   const float* __restrict__ Cadd,
                   const float* __restrict__ bias,
                   float* __restrict__ Draw,
                   float* __restrict__ Drelu,
                   int M, int N, int K)
{
    const int wave = (blockIdx.x * blockDim.x + threadIdx.x) >> 5;
    const int lane = threadIdx.x & 31;
    const int ntiles = N >> 6;                 // 64-wide N tiles
    const int total  = (M >> 5) * ntiles;      // 32-tall M tiles
    if (wave >= total) return;

    const int m0 = (wave / ntiles) << 5;
    const int n0 = (wave % ntiles) << 6;

    const int row  = lane & 15;                // M row (A) / N col (B, C/D)
    const int kgrp = (lane >> 4) << 1;         // 0 or 2: which K pair this lane holds

    v8f acc0[4], acc1[4];
#pragma unroll
    for (int t = 0; t < 4; ++t) {
        acc0[t] = (v8f){0.f,0.f,0.f,0.f,0.f,0.f,0.f,0.f};
        acc1[t] = (v8f){0.f,0.f,0.f,0.f,0.f,0.f,0.f,0.f};
    }

    const float* Arow0 = A + (size_t)(m0 + row) * lda + kgrp;        // m-half 0
    const float* Arow1 = Arow0 + (size_t)16 * lda;                    // m-half 1
    const float* Wp    = W + (size_t)kgrp * ldw + n0 + row;           // B base

    const int Kmain = K & ~3;

    // ---- branch-free main loop over full K4 steps ----
    for (int k0 = 0; k0 < Kmain; k0 += 4) {
        v2f a0, a1;
        a0[0] = Arow0[0];  a0[1] = Arow0[1];
        a1[0] = Arow1[0];  a1[1] = Arow1[1];
        Arow0 += 4;  Arow1 += 4;

        v2f b[4];
#pragma unroll
        for (int t = 0; t < 4; ++t) {
            b[t][0] = Wp[(size_t)0   * ldw + (t << 4)];
            b[t][1] = Wp[(size_t)1   * ldw + (t << 4)];
        }
        Wp += (size_t)4 * ldw;

#pragma unroll
        for (int t = 0; t < 4; ++t) {
            acc0[t] = __builtin_amdgcn_wmma_f32_16x16x4_f32(
                false, a0, false, b[t], (short)0, acc0[t], false, false);
            acc1[t] = __builtin_amdgcn_wmma_f32_16x16x4_f32(
                false, a1, false, b[t], (short)0, acc1[t], false, false);
        }
    }

    // ---- K tail (K%4 != 0): clamped-address loads + value select ----
    if (Kmain < K) {
        const int ka  = Kmain + kgrp;
        const int c0  = (ka     < K) ? ka     : (K - 1);  // safe index
        const int c1  = (ka + 1 < K) ? ka + 1 : (K - 1);
        const float z0 = (ka     < K) ? 1.f : 0.f;
        const float z1 = (ka + 1 < K) ? 1.f : 0.f;
        const float* Ab0 = A + (size_t)(m0 + row) * lda;
        const float* Ab1 = Ab0 + (size_t)16 * lda;

        v2f a0, a1;
        a0[0] = Ab0[c0] * z0;  a0[1] = Ab0[c1] * z1;
        a1[0] = Ab1[c0] * z0;  a1[1] = Ab1[c1] * z1;

        v2f b[4];
#pragma unroll
        for (int t = 0; t < 4; ++t) {
            b[t][0] = W[(size_t)c0 * ldw + n0 + (t << 4) + row] * z0;
            b[t][1] = W[(size_t)c1 * ldw + n0 + (t << 4) + row] * z1;
        }
#pragma unroll
        for (int t = 0; t < 4; ++t) {
            acc0[t] = __builtin_amdgcn_wmma_f32_16x16x4_f32(
                false, a0, false, b[t], (short)0, acc0[t], false, false);
            acc1[t] = __builtin_amdgcn_wmma_f32_16x16x4_f32(
                false, a1, false, b[t], (short)0, acc1[t], false, false);
        }
    }

    // ---- Epilogue. C/D layout: vgpr r, lanes 0-15 -> M=r ; lanes 16-31 -> M=8+r
    const int r0 = (lane >> 4) << 3;
#pragma unroll
    for (int half = 0; half < 2; ++half) {
        const v8f* acc = half ? acc1 : acc0;
        const int mh = m0 + (half << 4) + r0;
#pragma unroll
        for (int t = 0; t < 4; ++t) {
            const int n = n0 + (t << 4) + row;
            const float bn = bias ? bias[n] : 0.f;
#pragma unroll
            for (int r = 0; r < 8; ++r) {
                const int m = mh + r;
                float v = acc[t][r] + bn;
                if (Cadd)  v += Cadd[(size_t)m * N + n];
                if (Draw)  Draw[(size_t)m * N + n]  = v;
                if (Drelu) Drelu[(size_t)m * N + n] = fmaxf(v, 0.f);
            }
        }
    }
}

// ---------------------------------------------------------------------------
// a_msg[a,h] = sum_{j<6} msg[a2b[a][j], h]
// ---------------------------------------------------------------------------
__global__ __launch_bounds__(256)
void agg_kernel(const float* __restrict__ msg, const int* __restrict__ a2b,
                float* __restrict__ amsg, int n_atoms)
{
    int t = blockIdx.x * blockDim.x + threadIdx.x;
    if (t >= n_atoms * H) return;
    const int a = t >> 9;          // /512
    const int h = t & (H - 1);
    const int* nb = a2b + a * MAX_NB;
    float s = 0.f;
#pragma unroll
    for (int j = 0; j < MAX_NB; ++j)
        s += msg[(size_t)nb[j] * H + h];
    amsg[t] = s;
}

// ---------------------------------------------------------------------------
// pre[b,h] = amsg[b2a[b], h] - msg[b2revb[b], h]
// ---------------------------------------------------------------------------
__global__ __launch_bounds__(256)
void bondpre_kernel(const float* __restrict__ amsg, const float* __restrict__ msg,
                    const int* __restrict__ b2a, const int* __restrict__ b2revb,
                    float* __restrict__ pre, int n_bonds)
{
    int t = blockIdx.x * blockDim.x + threadIdx.x;
    if (t >= n_bonds * H) return;
    const int bb = t >> 9;
    const int h  = t & (H - 1);
    pre[t] = amsg[(size_t)b2a[bb] * H + h] - msg[(size_t)b2revb[bb] * H + h];
}

// ---------------------------------------------------------------------------
// score[p] = b_a2 + sum_c W_a2[c] * relu(U[b,i,c] + V[b,j,c])
// (b_a1 already folded into U). One wave32 per pair.
// ---------------------------------------------------------------------------
__global__ __launch_bounds__(256)
void score_kernel(const float* __restrict__ U, const float* __restrict__ V,
                  const float* __restrict__ W_a2, const float* __restrict__ b_a2,
                  float* __restrict__ scores)
{
    const int wave = (blockIdx.x * blockDim.x + threadIdx.x) >> 5;
    const int lane = threadIdx.x & 31;
    if (wave >= NPAIR) return;
    const int b   = wave / (A_SV * A_SU);
    const int rem = wave % (A_SV * A_SU);
    const int i   = rem >> 5;      // /A_SU
    const int j   = rem & 31;
    const float* u = U + (size_t)(b * A_SV + i) * (2 * H);
    const float* v = V + (size_t)(b * A_SU + j) * (2 * H);
    float s = 0.f;
    for (int c = lane; c < 2 * H; c += 32)
        s += W_a2[c] * fmaxf(u[c] + v[c], 0.f);
#pragma unroll
    for (int off = 16; off > 0; off >>= 1)
        s += __shfl_xor(s, off, 32);
    if (lane == 0) scores[wave] = s + b_a2[0];
}

// ---------------------------------------------------------------------------
// Global softmax stats over all NPAIR scores: red[0]=max, red[1]=sum(exp(s-max))
// ---------------------------------------------------------------------------
__global__ __launch_bounds__(1024)
void softmax_stats_kernel(const float* __restrict__ scores, int n,
                          float* __restrict__ red)
{
    __shared__ float sm[1024];
    const int t = threadIdx.x;
    float mx = -INFINITY;
    for (int idx = t; idx < n; idx += 1024) mx = fmaxf(mx, scores[idx]);
    sm[t] = mx; __syncthreads();
    for (int s = 512; s > 0; s >>= 1) {
        if (t < s) sm[t] = fmaxf(sm[t], sm[t + s]);
        __syncthreads();
    }
    const float gmax = sm[0]; __syncthreads();
    float sum = 0.f;
    for (int idx = t; idx < n; idx += 1024) sum += expf(scores[idx] - gmax);
    sm[t] = sum; __syncthreads();
    for (int s = 512; s > 0; s >>= 1) {
        if (t < s) sm[t] += sm[t + s];
        __syncthreads();
    }
    if (t == 0) { red[0] = gmax; red[1] = sm[0]; }
}

// ---------------------------------------------------------------------------
// w_sv[b,i] = sum_j att[b,i,j] ; w_su[b,j] = sum_i att[b,i,j]
// ---------------------------------------------------------------------------
__global__ __launch_bounds__(256)
void weights_kernel(const float* __restrict__ scores, const float* __restrict__ red,
                    float* __restrict__ w_sv, float* __restrict__ w_su)
{
    const int t = blockIdx.x * blockDim.x + threadIdx.x;
    const float mx = red[0];
    const float inv = 1.f / red[1];
    if (t < N_A_SV) {
        const float* s = scores + (size_t)t * A_SU;   // t = b*A_SV + i
        float acc = 0.f;
#pragma unroll
        for (int j = 0; j < A_SU; ++j) acc += expf(s[j] - mx);
        w_sv[t] = acc * inv;
    } else if (t < N_A_SV + N_A_SU) {
        const int u = t - N_A_SV;                     // u = b*A_SU + j
        const int b = u >> 5, j = u & 31;
        const float* s = scores + (size_t)b * A_SV * A_SU + j;
        float acc = 0.f;
#pragma unroll
        for (int i = 0; i < A_SV; ++i) acc += expf(s[i * A_SU] - mx);
        w_su[u] = acc * inv;
    }
}

// ---------------------------------------------------------------------------
// out[b, 0:512]     = mean_i  ah_sv[b,i,:]
// out[b, 512:1024]  = (1/768) sum_i  ah_sv[b,i,:] * w_sv[b,i]
// out[b, 1024:1536] = (1/768) sum_j  ah_su[b,j,:] * w_su[b,j]
// out[b, 1536:2048] = mean_j  ah_su[b,j,:]
// ---------------------------------------------------------------------------
__global__ __launch_bounds__(256)
void output_kernel(const float* __restrict__ ah_sv, const float* __restrict__ ah_su,
                   const float* __restrict__ w_sv, const float* __restrict__ w_su,
                   float* __restrict__ out)
{
    const int t = blockIdx.x * blockDim.x + threadIdx.x;
    if (t >= NB * 4 * H) return;
    const int b = t >> 11;         // /2048
    const int col = t & 2047;
    float val;
    if (col < H) {
        const float* p = ah_sv + (size_t)b * A_SV * H + col;
        float s = 0.f;
#pragma unroll
        for (int i = 0; i < A_SV; ++i) s += p[i * H];
        val = s * (1.f / A_SV);
    } else if (col < 2 * H) {
        const int c = col - H;
        const float* p = ah_sv + (size_t)b * A_SV * H + c;
        const float* w = w_sv + b * A_SV;
        float s = 0.f;
#pragma unroll
        for (int i = 0; i < A_SV; ++i) s += p[i * H] * w[i];
        val = s * (1.f / (A_SV * A_SU));
    } else if (col < 3 * H) {
        const int c = col - 2 * H;
        const float* p = ah_su + (size_t)b * A_SU * H + c;
        const float* w = w_su + b * A_SU;
        float s = 0.f;
#pragma unroll
        for (int j = 0; j < A_SU; ++j) s += p[j * H] * w[j];
        val = s * (1.f / (A_SV * A_SU));
    } else {
        const int c = col - 3 * H;
        const float* p = ah_su + (size_t)b * A_SU * H + c;
        float s = 0.f;
#pragma unroll
        for (int j = 0; j < A_SU; ++j) s += p[j * H];
        val = s * (1.f / A_SU);
    }
    out[t] = val;
}

// ---------------------------------------------------------------------------
// Host-side orchestration
// ---------------------------------------------------------------------------
static inline void launch_gemm(const float* A, int lda, const float* W, int ldw,
                               const float* add, const float* bias,
                               float* Draw, float* Drelu,
                               int M, int N, int K, hipStream_t s)
{
    const int waves  = (M >> 5) * (N >> 6);    // 32x64 tile per wave
    const int blocks = (waves + 7) / 8;        // 8 waves / 256-thread block
    gemm_wmma_f32<<<blocks, 256, 0, s>>>(A, lda, W, ldw, add, bias, Draw, Drelu, M, N, K);
}

extern "C" void kernel_launch(void* const* d_in, const int* in_sizes, int n_in,
                              void* d_out, int out_size, void* d_ws, size_t ws_size,
                              hipStream_t stream)
{
    (void)in_sizes; (void)n_in; (void)out_size; (void)ws_size;
    const float* f_atoms_sv = (const float*)d_in[0];
    const float* f_bonds_sv = (const float*)d_in[1];
    const int*   a2b_sv     = (const int*)d_in[2];
    const int*   b2a_sv     = (const int*)d_in[3];
    const int*   b2revb_sv  = (const int*)d_in[4];
    const float* f_atoms_su = (const float*)d_in[5];
    const float* f_bonds_su = (const float*)d_in[6];
    const int*   a2b_su     = (const int*)d_in[7];
    const int*   b2a_su     = (const int*)d_in[8];
    const int*   b2revb_su  = (const int*)d_in[9];
    const float* W_i  = (const float*)d_in[10];
    const float* W_h  = (const float*)d_in[11];
    const float* W_o  = (const float*)d_in[12];
    const float* b_o  = (const float*)d_in[13];
    const float* W_a1 = (const float*)d_in[14];
    const float* b_a1 = (const float*)d_in[15];
    const float* W_a2 = (const float*)d_in[16];
    const float* b_a2 = (const float*)d_in[17];
    float* out = (float*)d_out;

    // Workspace layout (floats)
    float* ws = (float*)d_ws;
    size_t o = 0;
    float* inp_sv   = ws + o; o += (size_t)N_B_SV * H;   // later reused as U (1536x1024)
    float* msg_sv   = ws + o; o += (size_t)N_B_SV * H;
    float* tmp_sv   = ws + o; o += (size_t)N_B_SV * H;   // bond "pre" buffer / W_o partial
    float* inp_su   = ws + o; o += (size_t)N_B_SU * H;   // later reused as V (2048x1024)
    float* msg_su   = ws + o; o += (size_t)N_B_SU * H;
    float* tmp_su   = ws + o; o += (size_t)N_B_SU * H;
    float* amsg     = ws + o; o += (size_t)N_A_SU * H;   // shared atom-message buffer
    float* atomh_sv = ws + o; o += (size_t)N_A_SV * H;
    float* atomh_su = ws + o; o += (size_t)N_A_SU * H;
    float* scores   = ws + o; o += (size_t)NPAIR;
    float* red      = ws + o; o += 2;
    float* w_sv     = ws + o; o += N_A_SV;
    float* w_su     = ws + o; o += N_A_SU;

    const int TB = 256;

    // ================= Solvent MPN =================
    // inp = f_bonds @ W_i ; msg = relu(inp)
    launch_gemm(f_bonds_sv, BOND_FDIM, W_i, H, nullptr, nullptr,
                inp_sv, msg_sv, N_B_SV, H, BOND_FDIM, stream);
    for (int it = 0; it < 2; ++it) {
        agg_kernel<<<(N_A_SV * H + TB - 1) / TB, TB, 0, stream>>>(msg_sv, a2b_sv, amsg, N_A_SV);
        bondpre_kernel<<<(N_B_SV * H + TB - 1) / TB, TB, 0, stream>>>(amsg, msg_sv, b2a_sv, b2revb_sv, tmp_sv, N_B_SV);
        // msg = relu(inp + pre @ W_h)
        launch_gemm(tmp_sv, H, W_h, H, inp_sv, nullptr,
                    nullptr, msg_sv, N_B_SV, H, H, stream);
    }
    agg_kernel<<<(N_A_SV * H + TB - 1) / TB, TB, 0, stream>>>(msg_sv, a2b_sv, amsg, N_A_SV);
    // atom_h = relu([f_atoms, a_msg] @ W_o + b_o)  (split-K, no concat buffer)
    launch_gemm(f_atoms_sv, ATOM_FDIM, W_o, H, nullptr, nullptr,
                tmp_sv, nullptr, N_A_SV, H, ATOM_FDIM, stream);
    launch_gemm(amsg, H, W_o + (size_t)ATOM_FDIM * H, H, tmp_sv, b_o,
                nullptr, atomh_sv, N_A_SV, H, H, stream);

    // ================= Solute MPN =================
    launch_gemm(f_bonds_su, BOND_FDIM, W_i, H, nullptr, nullptr,
                inp_su, msg_su, N_B_SU, H, BOND_FDIM, stream);
    for (int it = 0; it < 2; ++it) {
        agg_kernel<<<(N_A_SU * H + TB - 1) / TB, TB, 0, stream>>>(msg_su, a2b_su, amsg, N_A_SU);
        bondpre_kernel<<<(N_B_SU * H + TB - 1) / TB, TB, 0, stream>>>(amsg, msg_su, b2a_su, b2revb_su, tmp_su, N_B_SU);
        launch_gemm(tmp_su, H, W_h, H, inp_su, nullptr,
                    nullptr, msg_su, N_B_SU, H, H, stream);
    }
    agg_kernel<<<(N_A_SU * H + TB - 1) / TB, TB, 0, stream>>>(msg_su, a2b_su, amsg, N_A_SU);
    launch_gemm(f_atoms_su, ATOM_FDIM, W_o, H, nullptr, nullptr,
                tmp_su, nullptr, N_A_SU, H, ATOM_FDIM, stream);
    launch_gemm(amsg, H, W_o + (size_t)ATOM_FDIM * H, H, tmp_su, b_o,
                nullptr, atomh_su, N_A_SU, H, H, stream);

    // ================= Attention (factored) =================
    // U = atomh_sv @ W_a1[:H,:] + b_a1   (b_a1 folded here once)
    float* U = inp_sv;   // reuse: 1536*1024 == 3072*512
    float* V = inp_su;   // reuse: 2048*1024 == 4096*512
    launch_gemm(atomh_sv, H, W_a1, 2 * H, nullptr, b_a1,
                U, nullptr, N_A_SV, 2 * H, H, stream);
    // V = atomh_su @ W_a1[H:,:]
    launch_gemm(atomh_su, H, W_a1 + (size_t)H * 2 * H, 2 * H, nullptr, nullptr,
                V, nullptr, N_A_SU, 2 * H, H, stream);

    // scores: one wave per pair
    {
        const int waves = NPAIR;
        const int blocks = (waves + 7) / 8;
        score_kernel<<<blocks, TB, 0, stream>>>(U, V, W_a2, b_a2, scores);
    }
    softmax_stats_kernel<<<1, 1024, 0, stream>>>(scores, NPAIR, red);
    weights_kernel<<<(N_A_SV + N_A_SU + TB - 1) / TB, TB, 0, stream>>>(scores, red, w_sv, w_su);
    output_kernel<<<(NB * 4 * H + TB - 1) / TB, TB, 0, stream>>>(atomh_sv, atomh_su, w_sv, w_su, out);
}